// InterfacialDMI_50388556316705
// MI455X (gfx1250) — compile-verified
//
#include <hip/hip_runtime.h>
#include <hip/hip_bf16.h>

namespace {
constexpr int kNX = 512, kNY = 512, kNZ = 64;
constexpr int kNXYZ = kNX * kNY * kNZ;     // 16,777,216 cells
constexpr int kSI4  = (kNY * kNZ) / 4;     // i-neighbor stride, float4 units (8192)
constexpr int kSJ4  = kNZ / 4;             // j-neighbor stride, float4 units (16)
constexpr int kC4   = kNXYZ / 4;           // component stride, float4 units
constexpr int kBlk  = 256;                 // 8 waves per block on wave32
constexpr int kNQ   = kNXYZ / 4;           // 4,194,304 z-quads
constexpr int kNBlocks = kNQ / kBlk;       // 16384 block partials
}

typedef __attribute__((ext_vector_type(2))) float v2f;
typedef __attribute__((ext_vector_type(8))) float v8f;

// Pass 1: memory-bound stencil. 128-bit loads along contiguous z; zero BC via
// predicated loads; per-block LDS tree reduction -> one f32 partial per block.
__global__ __launch_bounds__(kBlk) void dmi_stencil(const float* __restrict__ x,
                                                    float* __restrict__ partials) {
  const int q  = blockIdx.x * kBlk + threadIdx.x;
  const int kq = q & (kNZ / 4 - 1);   // quad index along z
  const int r  = q >> 4;
  const int j  = r & (kNY - 1);
  const int i  = r >> 9;
  const int b4 = i * kSI4 + j * kSJ4 + kq;

  const float4* __restrict__ X = reinterpret_cast<const float4*>(x);
  const float4 zero4 = make_float4(0.f, 0.f, 0.f, 0.f);

  const float4 mx = X[b4];
  const float4 my = X[kC4 + b4];
  const float4 mz = X[2 * kC4 + b4];

  const bool hasL = (i > 0), hasR = (i < kNX - 1);
  const bool hasD = (j > 0), hasU = (j < kNY - 1);
  const float4 lx = hasL ? X[b4 - kSI4]           : zero4;
  const float4 lz = hasL ? X[2 * kC4 + b4 - kSI4] : zero4;
  const float4 rx = hasR ? X[b4 + kSI4]           : zero4;
  const float4 rz = hasR ? X[2 * kC4 + b4 + kSI4] : zero4;
  const float4 dy = hasD ? X[kC4 + b4 - kSJ4]     : zero4;
  const float4 dz = hasD ? X[2 * kC4 + b4 - kSJ4] : zero4;
  const float4 uy = hasU ? X[kC4 + b4 + kSJ4]     : zero4;
  const float4 uz = hasU ? X[2 * kC4 + b4 + kSJ4] : zero4;

  // E = mz*(lx - rx + dy - uy) + mx*(rz - lz) + my*(uz - dz), per z-lane
  float s;
  s  = mz.x * (lx.x - rx.x + dy.x - uy.x) + mx.x * (rz.x - lz.x) + my.x * (uz.x - dz.x);
  s += mz.y * (lx.y - rx.y + dy.y - uy.y) + mx.y * (rz.y - lz.y) + my.y * (uz.y - dz.y);
  s += mz.z * (lx.z - rx.z + dy.z - uy.z) + mx.z * (rz.z - lz.z) + my.z * (uz.z - dz.z);
  s += mz.w * (lx.w - rx.w + dy.w - uy.w) + mx.w * (rz.w - lz.w) + my.w * (uz.w - dz.w);

  __shared__ float red[kBlk];
  red[threadIdx.x] = s;
  __syncthreads();
#pragma unroll
  for (int off = kBlk / 2; off > 0; off >>= 1) {
    if (threadIdx.x < off) red[threadIdx.x] += red[threadIdx.x + off];
    __syncthreads();
  }
  if (threadIdx.x == 0) partials[blockIdx.x] = red[0];
}

// Pass 2: one wave32, EXEC all-1s. WMMA-based deterministic reduction of the
// 16384 block partials: A holds 64 fresh partials, B = ones, C accumulates
// 16 row sums (each row sum broadcast across its 16 columns).
__global__ __launch_bounds__(32) void dmi_reduce(const float* __restrict__ partials,
                                                 const float* __restrict__ Dparam,
                                                 float* __restrict__ out) {
  const int lane = threadIdx.x;  // 0..31, single wave
  v8f c = {};                    // 16x16 f32 accumulator (8 VGPRs)
  v2f ones;
  ones[0] = 1.0f;
  ones[1] = 1.0f;

  for (int base = 0; base < kNBlocks; base += 64) {
    v2f a;                       // A-matrix 16x4 f32: 2 VGPRs x 32 lanes = 64 values
    a[0] = partials[base + lane];
    a[1] = partials[base + 32 + lane];
    // D = A x ones + C  ->  each accumulator row gains the sum of 4 partials
    c = __builtin_amdgcn_wmma_f32_16x16x4_f32(
        /*neg_a=*/false, a, /*neg_b=*/false, ones,
        /*c_mod=*/(short)0, c, /*reuse_a=*/false, /*reuse_b=*/false);
  }

  // Per-lane: sum rows held in this lane's half (lanes 0-15: M=0..7, 16-31: M=8..15)
  float t = c[0] + c[1] + c[2] + c[3] + c[4] + c[5] + c[6] + c[7];
  // Combine the two half-wave row groups via lane permute (no LDS memory touched)
  float t2 = __int_as_float(
      __builtin_amdgcn_ds_bpermute((lane ^ 16) << 2, __float_as_int(t)));
  float total = t + t2;

  if (lane == 0) {
    const float scale = 0.5f * 4.0e-18f / (float)kNXYZ;  // 0.5*cellsize^2 / N
    out[0] = total * scale * Dparam[0];
  }
}

extern "C" void kernel_launch(void* const* d_in, const int* in_sizes, int n_in,
                              void* d_out, int out_size, void* d_ws, size_t ws_size,
                              hipStream_t stream) {
  (void)in_sizes; (void)n_in; (void)out_size; (void)ws_size;
  const float* x = (const float*)d_in[0];
  // d_in[1] = geo mask, unused by the forward (matches reference)
  const float* D = (const float*)d_in[2];
  float* partials = (float*)d_ws;          // 16384 floats = 64 KB scratch
  float* out = (float*)d_out;

  dmi_stencil<<<kNBlocks, kBlk, 0, stream>>>(x, partials);
  dmi_reduce<<<1, 32, 0, stream>>>(partials, D, out);
}